// MultipleNegativesRankingLoss_71021579206985
// MI455X (gfx1250) — compile-verified
//
#include <hip/hip_runtime.h>
#include <math.h>

// MultipleNegativesRankingLoss over energy distance, fp32 WMMA path for gfx1250.
// Shapes fixed by reference: x[N,L,D], y[M,D], mask[N,L]; out: scalar loss.
//
// v3: batched fragment loads (4xA + 16xB issued before the 16-WMMA chain so
// the scheduler can clause them and stagger s_wait_loadcnt), constant-offset
// addressing for the 4 B streams (imm24 offsets, no per-step 64-bit VALU
// adds), symmetric self-gram (upper tiles only, off-diagonal doubled), and
// v_max+v_sqrt safe-sqrt epilogue.

typedef __attribute__((ext_vector_type(2))) float v2f;
typedef __attribute__((ext_vector_type(8))) float v8f;

#define N_ 64
#define L_ 256
#define D_ 1024
#define M_ 128
#define SCALE_ 20.0f

#define TJSTRIDE (16 * D_)   // floats between adjacent 16-row j-tiles

__device__ __forceinline__ float wave_reduce_add(float v) {
#pragma unroll
  for (int off = 16; off > 0; off >>= 1) v += __shfl_down(v, off, 32);
  return v;
}

__device__ __forceinline__ float safe_sqrt(float d2) {
  // matches reference: sqrt(d2) for d2>0 else 0  (sqrt(0)==0)
  return __builtin_amdgcn_sqrtf(fmaxf(d2, 0.f));
}

// --- Kernel 0: row squared-norms for x (N*L rows) and y (M rows), one wave32 per row
__global__ void __launch_bounds__(256) norms_kernel(const float* __restrict__ x,
                                                    const float* __restrict__ y,
                                                    float* __restrict__ sqx,
                                                    float* __restrict__ sqy) {
  const int gw = blockIdx.x * 8 + (threadIdx.x >> 5);
  const int lane = threadIdx.x & 31;
  const int NROWS = N_ * L_ + M_;
  if (gw >= NROWS) return;
  const float* base = (gw < N_ * L_) ? (x + (size_t)gw * D_)
                                     : (y + (size_t)(gw - N_ * L_) * D_);
  float s = 0.f;
#pragma unroll
  for (int i = lane * 4; i < D_; i += 128) {
    float4 v = *(const float4*)(base + i);
    s += v.x * v.x + v.y * v.y + v.z * v.z + v.w * v.w;
  }
  s = wave_reduce_add(s);
  if (lane == 0) {
    if (gw < N_ * L_) sqx[gw] = s;
    else              sqy[gw - N_ * L_] = s;
  }
}

// --- Kernel 1: per-batch valid-token counts
__global__ void maskcnt_kernel(const int* __restrict__ mask, float* __restrict__ cntf) {
  const int n = threadIdx.x;
  if (n < N_) {
    int c = 0;
    for (int l = 0; l < L_; ++l) c += mask[n * L_ + l];
    cntf[n] = (float)c;
  }
}

// Shared inner block: accumulate 4 output tiles (1 A strip vs 4 B tiles) over
// one 16-float K-chunk. Loads are all issued before the WMMA chain.
__device__ __forceinline__ void mma_chunk(const float* __restrict__ pa,
                                          const float* __restrict__ pb0,
                                          int kk, v8f acc[4]) {
  v2f a[4];
  v2f b[4][4];
#pragma unroll
  for (int u = 0; u < 4; ++u)
    a[u] = *(const v2f*)(pa + kk + 4 * u);
#pragma unroll
  for (int jj = 0; jj < 4; ++jj)
#pragma unroll
    for (int u = 0; u < 4; ++u)
      b[jj][u] = *(const v2f*)(pb0 + jj * TJSTRIDE + kk + 4 * u);
#pragma unroll
  for (int u = 0; u < 4; ++u)
#pragma unroll
    for (int jj = 0; jj < 4; ++jj)
      acc[jj] = __builtin_amdgcn_wmma_f32_16x16x4_f32(false, a[u], false, b[jj][u],
                                                      (short)0, acc[jj],
                                                      false, false);
}

// --- Kernel 2: self energy distance ed_q[n] via fp32 WMMA gram tiles.
// One workgroup (8 waves) per n. Symmetric gram: 40 strips (ti, jg) with
// jg >= ti/4; tj > ti weighted 2x, tj == ti weighted 1x, tj < ti skipped
// (wave-uniform control -> EXEC stays all-1s around the WMMAs).
__global__ void __launch_bounds__(256) self_ed_kernel(const float* __restrict__ x,
                                                      const int* __restrict__ mask,
                                                      const float* __restrict__ sqx,
                                                      const float* __restrict__ cntf,
                                                      float* __restrict__ edq) {
  const int n = blockIdx.x;
  const int wave = threadIdx.x >> 5;
  const int lane = threadIdx.x & 31;
  const int lanel = lane & 15;   // row/col within 16-wide fragment
  const int laneh = lane >> 4;   // selects K-pair (A/B) / M-half (C)

  const float* xb = x + (size_t)n * L_ * D_;
  const float* sqb = sqx + n * L_;
  const int* mb = mask + n * L_;

  float sum = 0.f;
  for (int s = wave; s < 40; s += 8) {
    // Decode strip index -> (ti, jg), jg in [ti/4, 3]
    int rem = s, ti = 0, jg = 0;
    for (int r = 0; r < 16; ++r) {
      const int cnt = 4 - (r >> 2);
      if (rem < cnt) { ti = r; jg = (r >> 2) + rem; break; }
      rem -= cnt;
    }
    const int tjb = jg * 4;

    const float* pa = xb + (size_t)(ti * 16 + lanel) * D_ + laneh * 2;
    const float* pb0 = xb + (size_t)(tjb * 16 + lanel) * D_ + laneh * 2;

    v8f acc[4];
#pragma unroll
    for (int jj = 0; jj < 4; ++jj) acc[jj] = (v8f){};

#pragma unroll 1
    for (int kk = 0; kk < D_; kk += 16)
      mma_chunk(pa, pb0, kk, acc);

    // Row-side (shared across the 4 tiles of the strip)
    float sqr[8], mrow[8];
#pragma unroll
    for (int r = 0; r < 8; ++r) {
      const int row = ti * 16 + laneh * 8 + r;
      sqr[r] = sqb[row];
      mrow[r] = (float)mb[row];
    }

#pragma unroll
    for (int jj = 0; jj < 4; ++jj) {
      const int tj = tjb + jj;
      if (tj < ti) continue;                   // below diagonal: covered by symmetry
      const float w = (tj == ti) ? 1.f : 2.f;  // off-diagonal tiles count twice
      const int col = tj * 16 + lanel;
      const float sqc = sqb[col];
      const float mc = w * (float)mb[col];
#pragma unroll
      for (int r = 0; r < 8; ++r) {
        const float d = safe_sqrt(sqr[r] + sqc - 2.f * acc[jj][r]);
        sum += mc * mrow[r] * d;
      }
    }
  }

  __shared__ float sp[8];
  sum = wave_reduce_add(sum);
  if (lane == 0) sp[wave] = sum;
  __syncthreads();
  if (threadIdx.x == 0) {
    float tot = 0.f;
    for (int w = 0; w < 8; ++w) tot += sp[w];
    const float c = cntf[n];
    const float denom = fmaxf(c * c, 1.f);   // clip(sum(pair_mask), 1, None)
    edq[n] = tot / denom;
  }
}

// --- Kernel 3: cross energy distance + scores[n,m] = -SCALE*(2*ed_sums - ed_q)
// 16 L-tile-rows x 2 M-groups = 32 strips of 4 tiles; A fragment shared.
__global__ void __launch_bounds__(256) cross_ed_kernel(const float* __restrict__ x,
                                                       const float* __restrict__ y,
                                                       const int* __restrict__ mask,
                                                       const float* __restrict__ sqx,
                                                       const float* __restrict__ sqy,
                                                       const float* __restrict__ cntf,
                                                       const float* __restrict__ edq,
                                                       float* __restrict__ scores) {
  const int n = blockIdx.x;
  const int wave = threadIdx.x >> 5;
  const int lane = threadIdx.x & 31;
  const int lanel = lane & 15;
  const int laneh = lane >> 4;

  __shared__ float sed[M_];
  if (threadIdx.x < M_) sed[threadIdx.x] = 0.f;
  __syncthreads();

  const float* xb = x + (size_t)n * L_ * D_;
  const float* sqb = sqx + n * L_;
  const int* mb = mask + n * L_;

  for (int s = wave; s < 32; s += 8) {
    const int ti = s >> 1;
    const int tjb = (s & 1) * 4;

    const float* pa = xb + (size_t)(ti * 16 + lanel) * D_ + laneh * 2;
    const float* pb0 = y + (size_t)(tjb * 16 + lanel) * D_ + laneh * 2;

    v8f acc[4];
#pragma unroll
    for (int jj = 0; jj < 4; ++jj) acc[jj] = (v8f){};

#pragma unroll 1
    for (int kk = 0; kk < D_; kk += 16)
      mma_chunk(pa, pb0, kk, acc);

    float sqr[8], mrow[8];
#pragma unroll
    for (int r = 0; r < 8; ++r) {
      const int row = ti * 16 + laneh * 8 + r;
      sqr[r] = sqb[row];
      mrow[r] = (float)mb[row];
    }

#pragma unroll
    for (int jj = 0; jj < 4; ++jj) {
      const int m = (tjb + jj) * 16 + lanel;   // column index (same for all 8 regs)
      const float sqm = sqy[m];
      float csum = 0.f;
#pragma unroll
      for (int r = 0; r < 8; ++r) {
        const float d = safe_sqrt(sqr[r] + sqm - 2.f * acc[jj][r]);
        csum += mrow[r] * d;
      }
      atomicAdd(&sed[m], csum);                // LDS ds_add_f32
    }
  }
  __syncthreads();

  if (threadIdx.x < M_) {
    const float c = fmaxf(cntf[n], 1.f);
    const float eds = 2.f * (sed[threadIdx.x] / c) - edq[n];
    scores[n * M_ + threadIdx.x] = -SCALE_ * eds;
  }
}

// --- Kernel 4: log-softmax over M per row, diagonal NLL mean -> scalar loss
__global__ void __launch_bounds__(128) loss_kernel(const float* __restrict__ scores,
                                                   float* __restrict__ out) {
  const int tid = threadIdx.x;
  __shared__ float red[M_];
  float loss = 0.f;
  for (int n = 0; n < N_; ++n) {
    const float s = scores[n * M_ + tid];
    red[tid] = s;
    __syncthreads();
#pragma unroll
    for (int off = 64; off > 0; off >>= 1) {
      if (tid < off) red[tid] = fmaxf(red[tid], red[tid + off]);
      __syncthreads();
    }
    const float mx = red[0];
    __syncthreads();
    red[tid] = expf(s - mx);
    __syncthreads();
#pragma unroll
    for (int off = 64; off > 0; off >>= 1) {
      if (tid < off) red[tid] += red[tid + off];
      __syncthreads();
    }
    if (tid == 0) {
      const float lse = mx + logf(red[0]);
      loss += lse - scores[n * M_ + n];     // -logp[n,n]
    }
    __syncthreads();
  }
  if (tid == 0) *out = loss / (float)N_;
}

extern "C" void kernel_launch(void* const* d_in, const int* in_sizes, int n_in,
                              void* d_out, int out_size, void* d_ws, size_t ws_size,
                              hipStream_t stream) {
  (void)in_sizes; (void)n_in; (void)out_size; (void)ws_size;
  const float* x = (const float*)d_in[0];
  const float* y = (const float*)d_in[1];
  const int* mask = (const int*)d_in[2];
  float* out = (float*)d_out;

  float* ws = (float*)d_ws;
  float* sqx    = ws;                 // N*L = 16384
  float* sqy    = sqx + N_ * L_;      // M   = 128
  float* cntf   = sqy + M_;           // N   = 64
  float* edq    = cntf + N_;          // N   = 64
  float* scores = edq + N_;           // N*M = 8192

  const int rows = N_ * L_ + M_;
  const int nb = (rows + 7) / 8;      // 8 waves per 256-thread block
  hipLaunchKernelGGL(norms_kernel,   dim3(nb), dim3(256), 0, stream, x, y, sqx, sqy);
  hipLaunchKernelGGL(maskcnt_kernel, dim3(1),  dim3(64),  0, stream, mask, cntf);
  hipLaunchKernelGGL(self_ed_kernel, dim3(N_), dim3(256), 0, stream, x, mask, sqx, cntf, edq);
  hipLaunchKernelGGL(cross_ed_kernel,dim3(N_), dim3(256), 0, stream, x, y, mask, sqx, sqy, cntf, edq, scores);
  hipLaunchKernelGGL(loss_kernel,    dim3(1),  dim3(M_),  0, stream, scores, out);
}